// _TTA_65953517797577
// MI455X (gfx1250) — compile-verified
//
#include <hip/hip_runtime.h>
#include <stdint.h>

#define N_TTA   4
#define NBOX    3072
#define MTOT    (N_TTA * NBOX)     // 12288
#define NMS_T   0.5f
#define SEGS    8
#define SEGROWS (MTOT / SEGS)      // 1536
#define TILE    128

// ---------------------------------------------------------------------------
// gfx1250 async-tensor path helpers: DMA global -> LDS (no VGPR round trip),
// tracked with ASYNCcnt, drained with s_wait_asynccnt. Inline asm per
// CDNA5_HIP.md guidance (portable across both toolchains).
// ---------------------------------------------------------------------------
typedef __attribute__((address_space(3))) void lds_void_t;

__device__ __forceinline__ uint32_t lds_off(void* p) {
  // generic -> LDS addrspace cast truncates to the 32-bit LDS offset
  return (uint32_t)(uintptr_t)(lds_void_t*)p;
}

__device__ __forceinline__ void async_ld_b128(uint32_t lds, const void* sbase, uint32_t voff) {
  asm volatile("global_load_async_to_lds_b128 %0, %1, %2"
               :: "v"(lds), "v"(voff), "s"(sbase) : "memory");
}

__device__ __forceinline__ void async_ld_b32(uint32_t lds, const void* sbase, uint32_t voff) {
  asm volatile("global_load_async_to_lds_b32 %0, %1, %2"
               :: "v"(lds), "v"(voff), "s"(sbase) : "memory");
}

__device__ __forceinline__ void wait_async0() {
  asm volatile("s_wait_asynccnt 0x0" ::: "memory");
}

// ---------------------------------------------------------------------------
// Kernel 1: preprocess. d[t][n] = [s, cx*s, cy*s, w*s, h*s] with TTA flips,
// s = score/4. Seed buf/corners/area/count from round 0.
// ---------------------------------------------------------------------------
__global__ __launch_bounds__(256) void k_prep(const float* __restrict__ y,
                                              float* __restrict__ d,
                                              float* __restrict__ buf,
                                              float4* __restrict__ c4,
                                              float* __restrict__ ar,
                                              int* __restrict__ cnt) {
  int tid = blockIdx.x * blockDim.x + threadIdx.x;
  if (tid >= MTOT) return;
  int t = tid / NBOX;
  int n = tid % NBOX;
  const float* yr = y + (size_t)tid * 5;
  float s  = yr[0] * 0.25f;               // /T, T=4 exact
  float cx = yr[1], cy = yr[2];
  if (t == 1 || t == 3) cx = 1.0f - cx;
  if (t == 2 || t == 3) cy = 1.0f - cy;
  float w = yr[3], h = yr[4];
  float* dr = d + (size_t)tid * 5;
  dr[0] = s; dr[1] = cx * s; dr[2] = cy * s; dr[3] = w * s; dr[4] = h * s;
  if (t == 0) {
    float* b = buf + (size_t)n * 5;
    b[0] = dr[0]; b[1] = dr[1]; b[2] = dr[2]; b[3] = dr[3]; b[4] = dr[4];
    // corners exactly as reference: m_box = buf[:,1:]/buf[:,:1]
    float icx = dr[1] / s, icy = dr[2] / s, iw = dr[3] / s, ih = dr[4] / s;
    c4[n] = make_float4(icx - iw * 0.5f, icy - ih * 0.5f,
                        icx + iw * 0.5f, icy + ih * 0.5f);
    ar[n] = iw * ih;
  }
  if (tid == 0) *cnt = NBOX;
}

// ---------------------------------------------------------------------------
// Kernel 2: IoU + segment-local argmax. Grid: (NBOX/256 col blocks, SEGS row
// segments). Buffer corner tiles (128 rows) are DMA'd into LDS via
// global_load_async_to_lds_b128/b32, double-buffered, load(k+1) overlapped
// with compute(k).
// ---------------------------------------------------------------------------
__global__ __launch_bounds__(256) void k_iou_seg(const float* __restrict__ d,
                                                 const float4* __restrict__ c4,
                                                 const float* __restrict__ ar,
                                                 const int* __restrict__ cnt,
                                                 float* __restrict__ pmax,
                                                 int* __restrict__ pidx,
                                                 int t) {
  __shared__ float4 sc4[2][TILE];
  __shared__ float  sar[2][TILE];

  const int j   = blockIdx.x * blockDim.x + threadIdx.x;   // candidate column
  const int seg = blockIdx.y;
  const int count = *cnt;
  const int rbeg = seg * SEGROWS;
  const int rend = min(rbeg + SEGROWS, count);
  const int nrows = rend - rbeg;

  // my candidate box: o_box = others[:,1:]/others[:,:1]
  const float* o = d + (size_t)(t * NBOX + j) * 5;
  float s   = o[0];
  float ocx = o[1] / s, ocy = o[2] / s, ow = o[3] / s, oh = o[4] / s;
  float bx1 = ocx - ow * 0.5f, bx2 = ocx + ow * 0.5f;
  float by1 = ocy - oh * 0.5f, by2 = ocy + oh * 0.5f;
  float area_b = ow * oh;

  float best = -3.0f;   // below the reference's -1 "invalid" fill
  int   besti = 0;

  if (nrows > 0) {                       // uniform per block
    const int ntiles = (nrows + TILE - 1) / TILE;

    auto issue = [&](int k, int sel) {
      int base = rbeg + k * TILE;
      int rows = min(TILE, rend - base);
      int tid  = threadIdx.x;
      if (tid < rows) {
        async_ld_b128(lds_off(&sc4[sel][tid]), c4, (uint32_t)((base + tid) * 16));
      } else if (tid >= TILE && (tid - TILE) < rows) {
        int r = tid - TILE;
        async_ld_b32(lds_off(&sar[sel][r]), ar, (uint32_t)((base + r) * 4));
      }
    };

    issue(0, 0);
    for (int k = 0; k < ntiles; ++k) {
      int sel = k & 1;
      wait_async0();          // each wave drains its own ASYNCcnt
      __syncthreads();        // tile k visible to whole workgroup
      if (k + 1 < ntiles) issue(k + 1, sel ^ 1);   // overlap with compute
      int base = rbeg + k * TILE;
      int rows = min(TILE, rend - base);
      for (int r = 0; r < rows; ++r) {
        float4 c = sc4[sel][r];                     // (x1,y1,x2,y2) broadcast
        float iw = fminf(c.z, bx2) - fmaxf(c.x, bx1);
        float ih = fminf(c.w, by2) - fmaxf(c.y, by1);
        iw = fmaxf(iw, 0.0f);
        ih = fmaxf(ih, 0.0f);
        float inter = iw * ih;
        float uni   = sar[sel][r] + area_b - inter;
        float iou   = inter / fmaxf(uni, 1e-8f);
        if (iou > best) { best = iou; besti = base + r; }  // first-max ties
      }
      __syncthreads();        // compute(k) done before tile k+2 overwrites sel
    }
  }
  pmax[seg * NBOX + j] = best;
  pidx[seg * NBOX + j] = besti;
}

// ---------------------------------------------------------------------------
// Kernel 3: combine segment partials. Ascending segments + strict '>' keeps
// the lowest global row index on ties (matches jnp.argmax).
// ---------------------------------------------------------------------------
__global__ __launch_bounds__(256) void k_reduce(const float* __restrict__ pmax,
                                                const int* __restrict__ pidx,
                                                float* __restrict__ ioumax,
                                                int* __restrict__ idx) {
  int j = blockIdx.x * blockDim.x + threadIdx.x;
  if (j >= NBOX) return;
  float best = -4.0f; int bi = 0;
  for (int s = 0; s < SEGS; ++s) {
    float v = pmax[s * NBOX + j];
    if (v > best) { best = v; bi = pidx[s * NBOX + j]; }
  }
  ioumax[j] = best;
  idx[j]    = bi;
}

// ---------------------------------------------------------------------------
// Kernel 4: deterministic scatter-add via gather. For each live buffer row i,
// sum all matched candidates whose argmax row == i (ascending j).
// ---------------------------------------------------------------------------
__global__ __launch_bounds__(256) void k_accum(float* __restrict__ buf,
                                               const float* __restrict__ d,
                                               const float* __restrict__ ioumax,
                                               const int* __restrict__ idx,
                                               const int* __restrict__ cnt,
                                               int t) {
  int i = blockIdx.x * blockDim.x + threadIdx.x;
  int count = *cnt;
  if (i >= count) return;
  float a0 = 0.f, a1 = 0.f, a2 = 0.f, a3 = 0.f, a4 = 0.f;
  bool any = false;
  for (int j = 0; j < NBOX; ++j) {            // uniform scalar-friendly scan
    if (ioumax[j] >= NMS_T && idx[j] == i) {
      const float* o = d + (size_t)(t * NBOX + j) * 5;
      a0 += o[0]; a1 += o[1]; a2 += o[2]; a3 += o[3]; a4 += o[4];
      any = true;
    }
  }
  if (any) {
    float* b = buf + (size_t)i * 5;
    b[0] += a0; b[1] += a1; b[2] += a2; b[3] += a3; b[4] += a4;
  }
}

// ---------------------------------------------------------------------------
// Kernel 5 (single block): append unmatched (block scan for positions in
// ascending j order, matching cumsum semantics), bump count, refresh
// corners/areas for all live rows.
// ---------------------------------------------------------------------------
__global__ __launch_bounds__(256) void k_append(float* __restrict__ buf,
                                                float4* __restrict__ c4,
                                                float* __restrict__ ar,
                                                const float* __restrict__ d,
                                                const float* __restrict__ ioumax,
                                                int* __restrict__ cnt,
                                                int t) {
  __shared__ int sc[256];
  __shared__ int s_count;
  const int tid = threadIdx.x;
  const int CH  = NBOX / 256;                 // 12 candidates per thread
  if (tid == 0) s_count = *cnt;
  __syncthreads();
  const int count = s_count;

  int local = 0;
  for (int e = 0; e < CH; ++e) {
    int j = tid * CH + e;
    if (!(ioumax[j] >= NMS_T)) local++;
  }
  sc[tid] = local;
  __syncthreads();
  // Hillis-Steele inclusive scan over 256 partials
  for (int off = 1; off < 256; off <<= 1) {
    int v = (tid >= off) ? sc[tid - off] : 0;
    __syncthreads();
    sc[tid] += v;
    __syncthreads();
  }
  const int excl  = sc[tid] - local;
  const int total = sc[255];

  int w = count + excl;
  for (int e = 0; e < CH; ++e) {
    int j = tid * CH + e;
    if (!(ioumax[j] >= NMS_T)) {
      const float* o = d + (size_t)(t * NBOX + j) * 5;
      float* b = buf + (size_t)w * 5;
      b[0] = o[0]; b[1] = o[1]; b[2] = o[2]; b[3] = o[3]; b[4] = o[4];
      w++;
    }
  }
  if (tid == 0) *cnt = count + total;
  __threadfence_block();
  __syncthreads();

  const int newc = count + total;
  for (int i = tid; i < newc; i += 256) {
    const float* b = buf + (size_t)i * 5;
    float s  = b[0];
    float cx = b[1] / s, cy = b[2] / s, bw = b[3] / s, bh = b[4] / s;
    c4[i] = make_float4(cx - bw * 0.5f, cy - bh * 0.5f,
                        cx + bw * 0.5f, cy + bh * 0.5f);
    ar[i] = bw * bh;
  }
}

// ---------------------------------------------------------------------------
// Kernel 6: finalize output (de-weight boxes, zero invalid rows).
// ---------------------------------------------------------------------------
__global__ __launch_bounds__(256) void k_final(const float* __restrict__ buf,
                                               const int* __restrict__ cnt,
                                               float* __restrict__ out) {
  int i = blockIdx.x * blockDim.x + threadIdx.x;
  if (i >= MTOT) return;
  int count = *cnt;
  float* o = out + (size_t)i * 5;
  if (i < count) {
    const float* b = buf + (size_t)i * 5;
    float s = b[0];
    o[0] = s;
    o[1] = b[1] / s; o[2] = b[2] / s; o[3] = b[3] / s; o[4] = b[4] / s;
  } else {
    o[0] = 0.f; o[1] = 0.f; o[2] = 0.f; o[3] = 0.f; o[4] = 0.f;
  }
}

// ---------------------------------------------------------------------------
extern "C" void kernel_launch(void* const* d_in, const int* in_sizes, int n_in,
                              void* d_out, int out_size, void* d_ws, size_t ws_size,
                              hipStream_t stream) {
  (void)in_sizes; (void)n_in; (void)out_size; (void)ws_size;
  const float* y = (const float*)d_in[0];
  float* out = (float*)d_out;

  // carve scratch (16B aligned slices)
  char* p = (char*)d_ws;
  auto take = [&](size_t bytes) -> char* {
    char* r = p;
    p += (bytes + 15) & ~(size_t)15;
    return r;
  };
  float*  d_d    = (float*) take((size_t)MTOT * 5 * sizeof(float));
  float*  d_buf  = (float*) take((size_t)(MTOT + 1) * 5 * sizeof(float));
  float4* d_c4   = (float4*)take((size_t)MTOT * sizeof(float4));
  float*  d_ar   = (float*) take((size_t)MTOT * sizeof(float));
  float*  d_iou  = (float*) take((size_t)NBOX * sizeof(float));
  int*    d_idx  = (int*)   take((size_t)NBOX * sizeof(int));
  float*  d_pmax = (float*) take((size_t)SEGS * NBOX * sizeof(float));
  int*    d_pidx = (int*)   take((size_t)SEGS * NBOX * sizeof(int));
  int*    d_cnt  = (int*)   take(sizeof(int));

  k_prep<<<MTOT / 256, 256, 0, stream>>>(y, d_d, d_buf, d_c4, d_ar, d_cnt);

  for (int t = 1; t < N_TTA; ++t) {
    dim3 gIoU(NBOX / 256, SEGS);
    k_iou_seg<<<gIoU, 256, 0, stream>>>(d_d, d_c4, d_ar, d_cnt, d_pmax, d_pidx, t);
    k_reduce<<<NBOX / 256, 256, 0, stream>>>(d_pmax, d_pidx, d_iou, d_idx);
    k_accum<<<MTOT / 256, 256, 0, stream>>>(d_buf, d_d, d_iou, d_idx, d_cnt, t);
    k_append<<<1, 256, 0, stream>>>(d_buf, d_c4, d_ar, d_d, d_iou, d_cnt, t);
  }

  k_final<<<MTOT / 256, 256, 0, stream>>>(d_buf, d_cnt, out);
}